// DualAttention_26895085207699
// MI455X (gfx1250) — compile-verified
//
#include <hip/hip_runtime.h>
#include <math.h>

// ---------------------------------------------------------------------------
// DualAttention for MI455X (gfx1250, wave32).
// All matmuls via V_WMMA_F32_16X16X4_F32 (f32 fidelity; workload is memory /
// transcendental bound, so lower-precision WMMA buys nothing measurable).
// GEMMs use 32x32 macro-tiles per wave: 4 wmma per 4 b64 loads.
// ---------------------------------------------------------------------------

typedef __attribute__((ext_vector_type(2))) float v2f;
typedef __attribute__((ext_vector_type(8))) float v8f;

#define WAVE 32
#define Bn   64
#define Ln   285
#define LP   288          // padded L (multiple of 32)
#define Lm1  284
#define Dd   256
#define IDIM 128
#define NITEM 99999       // emb_w rows 1..99999

// ---- WMMA wrapper ---------------------------------------------------------
__device__ __forceinline__ v8f wmma4(v2f a, v2f b, v8f c) {
  // (neg_a, A, neg_b, B, c_mod, C, reuse_a, reuse_b)
  return __builtin_amdgcn_wmma_f32_16x16x4_f32(false, a, false, b, (short)0, c,
                                               false, false);
}

// ---- wave32 reductions ----------------------------------------------------
__device__ __forceinline__ float wave_sum(float v) {
  for (int off = 16; off > 0; off >>= 1) v += __shfl_xor(v, off, 32);
  return v;
}
__device__ __forceinline__ float wave_max(float v) {
  for (int off = 16; off > 0; off >>= 1) v = fmaxf(v, __shfl_xor(v, off, 32));
  return v;
}

// ---------------------------------------------------------------------------
// WMMA-f32 GEMM, one wave -> one 32x32 C macro-tile (2x2 WMMA tiles).
//   C[b] = act( scale * A[b](M x K) * B'[b](K x N) + bias + resid )
//   TRANSB=1: B'[k][n] = B[n*ldb + k]   (X @ W^T, weight stored [N,K])
//   TRANSB=0: B'[k][n] = B[k*ldb + n]
// Mt2/Nt2 = M/32, N/32 macro-tile counts; K multiple of 4.
// Fragment layout per CDNA5 ISA 7.12.2:
//   A: lane m=lane&15, k = 2*(lane>>4)+v ; B mirrored with n=lane&15
//   C: row = r + 8*(lane>>4), col = lane&15
// ---------------------------------------------------------------------------
template <int TRANSB>
__global__ void k_gemm32(const float* __restrict__ A, const float* __restrict__ Bm,
                         float* __restrict__ C,
                         int Mt2, int Nt2, int K, int lda, int ldb, int ldc,
                         long long sA, long long sB, long long sC,
                         const float* __restrict__ bias, int act,
                         const float* __restrict__ resid, float scale) {
  const int wave = threadIdx.x / WAVE;
  const int lane = threadIdx.x % WAVE;
  const int tiles = Mt2 * Nt2;
  const int tile = blockIdx.x * (blockDim.x / WAVE) + wave;
  if (tile >= tiles) return;                       // wave-uniform guard
  const int b = blockIdx.y;
  const float* Ab = A + (long long)b * sA;
  const float* Bb = Bm + (long long)b * sB;
  float* Cb = C + (long long)b * sC;

  const int mt = tile / Nt2, nt = tile % Nt2;
  const int l15 = lane & 15;
  const int kb = (lane >> 4) * 2;
  const int m0 = mt * 32 + l15;
  const int n0 = nt * 32 + l15;

  const float* a0p = Ab + (long long)m0 * lda + kb;
  const float* a1p = a0p + 16 * lda;
  const float* b0p;
  const float* b1p;
  int bstep;
  if (TRANSB) {
    b0p = Bb + (long long)n0 * ldb + kb;
    b1p = b0p + 16 * ldb;
    bstep = 4;
  } else {
    b0p = Bb + (long long)kb * ldb + n0;
    b1p = b0p + 16;
    bstep = 4 * ldb;
  }

  v8f acc00 = {}, acc01 = {}, acc10 = {}, acc11 = {};
#pragma unroll 2
  for (int k0 = 0; k0 < K; k0 += 4) {
    v2f a0, a1, b0, b1;
    a0.x = a0p[0];  a0.y = a0p[1];
    a1.x = a1p[0];  a1.y = a1p[1];
    if (TRANSB) {
      b0.x = b0p[0];  b0.y = b0p[1];
      b1.x = b1p[0];  b1.y = b1p[1];
    } else {
      b0.x = b0p[0];  b0.y = b0p[ldb];
      b1.x = b1p[0];  b1.y = b1p[ldb];
    }
    acc00 = wmma4(a0, b0, acc00);
    acc01 = wmma4(a0, b1, acc01);
    acc10 = wmma4(a1, b0, acc10);
    acc11 = wmma4(a1, b1, acc11);
    a0p += 4;  a1p += 4;  b0p += bstep;  b1p += bstep;
  }

  const int c0 = nt * 32 + l15;
  const int c1 = c0 + 16;
  const float bv0 = bias ? bias[c0] : 0.0f;
  const float bv1 = bias ? bias[c1] : 0.0f;
  const int rbase = mt * 32 + ((lane >> 4) * 8);
  for (int r = 0; r < 8; ++r) {
    const int row0 = rbase + r;
    const int row1 = row0 + 16;
    float v00 = acc00[r] * scale + bv0;
    float v01 = acc01[r] * scale + bv1;
    float v10 = acc10[r] * scale + bv0;
    float v11 = acc11[r] * scale + bv1;
    if (resid) {
      v00 += resid[(long long)row0 * ldc + c0];
      v01 += resid[(long long)row0 * ldc + c1];
      v10 += resid[(long long)row1 * ldc + c0];
      v11 += resid[(long long)row1 * ldc + c1];
    }
    if (act == 1) {
      v00 = fmaxf(v00, 0.0f);  v01 = fmaxf(v01, 0.0f);
      v10 = fmaxf(v10, 0.0f);  v11 = fmaxf(v11, 0.0f);
    }
    Cb[(long long)row0 * ldc + c0] = v00;
    Cb[(long long)row0 * ldc + c1] = v01;
    Cb[(long long)row1 * ldc + c0] = v10;
    Cb[(long long)row1 * ldc + c1] = v11;
  }
}

// ---- embedding gather + concat into padded [B, LP, 256] -------------------
__global__ void k_embed(const int* __restrict__ x, const int* __restrict__ pos,
                        const float* __restrict__ emb,
                        const float* __restrict__ pemb, float* __restrict__ xp) {
  const int bl = blockIdx.x;            // 0 .. B*LP-1
  const int b = bl / LP, l = bl % LP;
  const int t = threadIdx.x;            // 0 .. 255
  float v = 0.0f;
  if (l < Ln) {
    if (t < IDIM)
      v = emb[(long long)x[b * Ln + l] * IDIM + t];
    else
      v = pemb[(long long)pos[b * Ln + l] * IDIM + (t - IDIM)];
  }
  xp[(long long)bl * Dd + t] = v;
}

// ---- per-batch alpha = add_value(sigmoid(row(L-1) . w + b) + 1) -----------
__global__ void k_alpha(const float* __restrict__ src,
                        const float* __restrict__ aw,
                        const float* __restrict__ ab, float* __restrict__ out) {
  __shared__ float red[256];
  const int b = blockIdx.x, t = threadIdx.x;
  red[t] = src[((long long)b * LP + (Ln - 1)) * Dd + t] * aw[t];
  __syncthreads();
  for (int s = 128; s > 0; s >>= 1) {
    if (t < s) red[t] += red[t + s];
    __syncthreads();
  }
  if (t == 0) {
    const float z = red[0] + ab[0];
    const float sg = 1.0f / (1.0f + __expf(-z));
    float a = sg + 1.0f;
    if (a == 1.0f) a = 1.0001f;
    out[b] = a;
  }
}

// ---- entmax bisection, one wave per row, 9 elements per lane --------------
__device__ __forceinline__ void entmax_row9(const float (&Xs)[9], float alpha,
                                            float dlen, float (&p)[9]) {
  const float am1 = alpha - 1.0f;
  const float inv = 1.0f / am1;
  float mx = -__builtin_inff();
  for (int j = 0; j < 9; ++j) mx = fmaxf(mx, Xs[j]);
  mx = wave_max(mx);
  float tau_lo = mx - 1.0f;
  const float tau_hi = mx - __expf(am1 * __logf(1.0f / dlen));
  float s = 0.0f;
  for (int j = 0; j < 9; ++j) {
    const float z = Xs[j] - tau_lo;
    s += (z > 0.0f) ? __expf(inv * __logf(z)) : 0.0f;
  }
  const float f_lo = wave_sum(s) - 1.0f;
  float dm = tau_hi - tau_lo;
  for (int it = 0; it < 50; ++it) {
    dm *= 0.5f;
    const float tau = tau_lo + dm;
    float sm = 0.0f;
    for (int j = 0; j < 9; ++j) {
      const float z = Xs[j] - tau;
      p[j] = (z > 0.0f) ? __expf(inv * __logf(z)) : 0.0f;
      sm += p[j];
    }
    const float fm = wave_sum(sm) - 1.0f;
    if (fm * f_lo >= 0.0f) tau_lo = tau;  // wave-uniform
  }
  float tot = 0.0f;
  for (int j = 0; j < 9; ++j) tot += p[j];
  tot = wave_sum(tot);
  const float invt = 1.0f / tot;
  for (int j = 0; j < 9; ++j) p[j] *= invt;
}

// entmax over score rows [B*LP, LP], in place; masked/padded cols -> 0.
__global__ void k_entmax_local(float* __restrict__ S, const int* __restrict__ x,
                               const float* __restrict__ aent) {
  const int wid = blockIdx.x * (blockDim.x / WAVE) + threadIdx.x / WAVE;
  const int lane = threadIdx.x % WAVE;
  if (wid >= Bn * LP) return;
  const int b = wid / LP, q = wid % LP;
  float* row = S + (long long)wid * LP;
  if (q >= Ln) {  // padded query rows -> zero so att@x_ K-loop stays exact
    for (int j = 0; j < LP / WAVE; ++j) row[lane + j * WAVE] = 0.0f;
    return;
  }
  const float alpha = aent[b];
  const float am1 = alpha - 1.0f;
  float Xs[9], p[9];
  for (int j = 0; j < 9; ++j) {
    const int e = lane + j * WAVE;
    const bool valid = (e < Ln) && (x[b * Ln + e] != 0);
    Xs[j] = valid ? row[e] * am1 : -__builtin_inff();
  }
  entmax_row9(Xs, alpha, (float)Ln, p);
  for (int j = 0; j < 9; ++j) row[lane + j * WAVE] = p[j];
}

// global entmax: one wave per batch over al[b, 0..283], in place.
__global__ void k_entmax_global(float* __restrict__ al, const int* __restrict__ x,
                                const float* __restrict__ aglob) {
  const int b = blockIdx.x * (blockDim.x / WAVE) + threadIdx.x / WAVE;
  const int lane = threadIdx.x % WAVE;
  if (b >= Bn) return;
  float* row = al + (long long)b * LP;
  const float alpha = aglob[b];
  const float am1 = alpha - 1.0f;
  float Xs[9], p[9];
  for (int j = 0; j < 9; ++j) {
    const int e = lane + j * WAVE;
    const bool valid = (e < Lm1) && (x[b * Ln + e] != 0);
    Xs[j] = valid ? row[e] * am1 : -__builtin_inff();
  }
  entmax_row9(Xs, alpha, (float)Lm1, p);
  for (int j = 0; j < 9; ++j) row[lane + j * WAVE] = p[j];
}

// ---- layernorm over 256-wide rows, in place -------------------------------
__global__ void k_ln(float* __restrict__ xio, const float* __restrict__ g,
                     const float* __restrict__ bta) {
  __shared__ float red[256];
  __shared__ float mv[2];
  const int t = threadIdx.x;
  float* r = xio + (long long)blockIdx.x * Dd;
  const float v = r[t];
  red[t] = v;
  __syncthreads();
  for (int s = 128; s > 0; s >>= 1) {
    if (t < s) red[t] += red[t + s];
    __syncthreads();
  }
  if (t == 0) mv[0] = red[0] * (1.0f / Dd);
  __syncthreads();
  const float d = v - mv[0];
  red[t] = d * d;
  __syncthreads();
  for (int s = 128; s > 0; s >>= 1) {
    if (t < s) red[t] += red[t + s];
    __syncthreads();
  }
  if (t == 0) mv[1] = rsqrtf(red[0] * (1.0f / Dd) + 1e-5f);
  __syncthreads();
  r[t] = d * mv[1] * g[t] + bta[t];
}

// ---- msw2[b,:] = m_s[b] @ atten_w2 ---------------------------------------
__global__ void k_msw2(const float* __restrict__ lnout,
                       const float* __restrict__ w2, float* __restrict__ msw2) {
  __shared__ float msv[Dd];
  const int b = blockIdx.x, n = threadIdx.x;
  msv[n] = lnout[((long long)b * LP + (Ln - 1)) * Dd + n];
  __syncthreads();
  float s = 0.0f;
  for (int k = 0; k < Dd; ++k) s += msv[k] * w2[k * Dd + n];
  msw2[b * Dd + n] = s;
}

// ---- al[b,k] = relu(hv[b,k]+msw2[b]+bias) . atten_w0  (one wave / (b,k)) --
__global__ void k_al(const float* __restrict__ hv, const float* __restrict__ msw2,
                     const float* __restrict__ abias, const float* __restrict__ w0,
                     float* __restrict__ al) {
  const int wid = blockIdx.x * (blockDim.x / WAVE) + threadIdx.x / WAVE;
  const int lane = threadIdx.x % WAVE;
  if (wid >= Bn * Lm1) return;
  const int b = wid / Lm1, k = wid % Lm1;
  const float* h = hv + ((long long)b * LP + k) * Dd;
  const float* mw = msw2 + b * Dd;
  float s = 0.0f;
  for (int j = 0; j < Dd / WAVE; ++j) {
    const int n = lane + j * WAVE;
    s += fmaxf(h[n] + mw[n] + abias[n], 0.0f) * w0[n];
  }
  s = wave_sum(s);
  if (lane == 0) al[b * LP + k] = s;
}

// ---- global_c[b,d] = sum_k aw[b,k] * x_s[b,k,d] ---------------------------
__global__ void k_globalc(const float* __restrict__ aw, const float* __restrict__ xp,
                          float* __restrict__ gc) {
  __shared__ float w[LP];
  const int b = blockIdx.x, d = threadIdx.x;
  for (int i = d; i < LP; i += Dd) w[i] = aw[b * LP + i];
  __syncthreads();
  float s = 0.0f;
  for (int k = 0; k < Lm1; ++k) s += w[k] * xp[((long long)b * LP + k) * Dd + d];
  gc[b * Dd + d] = s;
}

// ---- c = selu(concat(gc, m_s) @ w_f^T + b); l_c = c / ||c|| ---------------
__global__ void k_final(const float* __restrict__ gc, const float* __restrict__ lnout,
                        const float* __restrict__ wf, const float* __restrict__ bf,
                        float* __restrict__ lc) {
  __shared__ float in[2 * Dd];
  __shared__ float red[128];
  __shared__ float invn;
  const int b = blockIdx.x, o = threadIdx.x;  // 128 threads
  const float* ms = lnout + ((long long)b * LP + (Ln - 1)) * Dd;
  for (int i = o; i < Dd; i += 128) {
    in[i] = gc[b * Dd + i];
    in[Dd + i] = ms[i];
  }
  __syncthreads();
  float s = bf[o];
  for (int k = 0; k < 2 * Dd; ++k) s += wf[o * 2 * Dd + k] * in[k];
  const float scl = 1.0507009873554804934193349852946f;
  const float alp = 1.6732632423543772848170429916717f;
  const float c = (s > 0.0f) ? scl * s : scl * alp * (__expf(s) - 1.0f);
  red[o] = c * c;
  __syncthreads();
  for (int st = 64; st > 0; st >>= 1) {
    if (o < st) red[o] += red[o + st];
    __syncthreads();
  }
  if (o == 0) invn = rsqrtf(red[0]);
  __syncthreads();
  lc[b * IDIM + o] = c * invn;
}

// ---- inv_norm of emb rows 1..99999 (one wave per row) ---------------------
__global__ void k_invnorm(const float* __restrict__ emb, float* __restrict__ invn) {
  const int wid = blockIdx.x * (blockDim.x / WAVE) + threadIdx.x / WAVE;
  const int lane = threadIdx.x % WAVE;
  if (wid >= NITEM) return;
  const float* row = emb + (long long)(wid + 1) * IDIM;
  float s = 0.0f;
  for (int j = 0; j < IDIM / WAVE; ++j) {
    const float v = row[lane + j * WAVE];
    s += v * v;
  }
  s = wave_sum(s);
  if (lane == 0) invn[wid] = rsqrtf(s);
}

// ---- z[64, 99999] = 20 * l_c @ (emb[1:]/||.||)^T --------------------------
// One wave per 16-col tile computing all four 16-row M tiles; l_c staged in
// LDS (stride 132 -> all 32 lanes hit distinct banks on ds_load_b64).
#define LCP 132
__global__ void k_z(const float* __restrict__ lc, const float* __restrict__ emb,
                    const float* __restrict__ invn, float* __restrict__ z) {
  __shared__ float slc[64 * LCP];
  const int tid = threadIdx.x;
  for (int i = tid; i < 64 * IDIM; i += 256)
    slc[(i >> 7) * LCP + (i & 127)] = lc[i];
  __syncthreads();

  const int NT = (NITEM + 15) / 16;  // 6250 N tiles
  const int wid = blockIdx.x * 8 + (tid >> 5);
  const int lane = tid & 31;
  if (wid >= NT) return;  // wave-uniform
  const int n = wid * 16 + (lane & 15);
  const bool nv = n < NITEM;
  const int kb = (lane >> 4) * 2;
  const float inw = nv ? invn[n] : 0.0f;
  const float* er = emb + (long long)(nv ? n + 1 : 1) * IDIM + kb;
  // get the 512B embedding row streaming before the accumulate loop
  __builtin_prefetch(er, 0, 1);
  __builtin_prefetch(er + 64, 0, 1);

  const int aoff = (lane & 15) * LCP + kb;
  v8f acc0 = {}, acc1 = {}, acc2 = {}, acc3 = {};
#pragma unroll 4
  for (int k0 = 0; k0 < IDIM; k0 += 4) {
    v2f bb, a0, a1, a2, a3;
    bb.x = er[k0] * inw;
    bb.y = er[k0 + 1] * inw;
    a0.x = slc[aoff + k0];
    a0.y = slc[aoff + k0 + 1];
    a1.x = slc[aoff + 16 * LCP + k0];
    a1.y = slc[aoff + 16 * LCP + k0 + 1];
    a2.x = slc[aoff + 32 * LCP + k0];
    a2.y = slc[aoff + 32 * LCP + k0 + 1];
    a3.x = slc[aoff + 48 * LCP + k0];
    a3.y = slc[aoff + 48 * LCP + k0 + 1];
    acc0 = wmma4(a0, bb, acc0);
    acc1 = wmma4(a1, bb, acc1);
    acc2 = wmma4(a2, bb, acc2);
    acc3 = wmma4(a3, bb, acc3);
  }
  if (nv) {
    for (int r = 0; r < 8; ++r) {
      const int rr = r + ((lane >> 4) * 8);
      z[(long long)rr * NITEM + n]        = acc0[r] * 20.0f;
      z[(long long)(rr + 16) * NITEM + n] = acc1[r] * 20.0f;
      z[(long long)(rr + 32) * NITEM + n] = acc2[r] * 20.0f;
      z[(long long)(rr + 48) * NITEM + n] = acc3[r] * 20.0f;
    }
  }
}

// ---------------------------------------------------------------------------
extern "C" void kernel_launch(void* const* d_in, const int* in_sizes, int n_in,
                              void* d_out, int out_size, void* d_ws, size_t ws_size,
                              hipStream_t stream) {
  const int* x = (const int*)d_in[0];
  const int* pos = (const int*)d_in[1];
  const float* emb = (const float*)d_in[2];
  const float* pemb = (const float*)d_in[3];
  const float* atten_w0 = (const float*)d_in[4];
  const float* atten_w1 = (const float*)d_in[5];
  const float* atten_w2 = (const float*)d_in[6];
  const float* atten_bias = (const float*)d_in[7];
  const float* mlp_w = (const float*)d_in[8];
  const float* mlp_b = (const float*)d_in[9];
  const float* sa_w1_w = (const float*)d_in[10];
  const float* sa_w1_b = (const float*)d_in[11];
  const float* sa_w2_w = (const float*)d_in[12];
  const float* sa_w2_b = (const float*)d_in[13];
  const float* ln_g = (const float*)d_in[14];
  const float* ln_b = (const float*)d_in[15];
  const float* w_f_w = (const float*)d_in[16];
  const float* w_f_b = (const float*)d_in[17];
  const float* alpha_w = (const float*)d_in[18];
  const float* alpha_b = (const float*)d_in[19];

  float* ws = (float*)d_ws;
  const long long NLD = (long long)Bn * LP * Dd;  // 4,718,592
  const long long NLL = (long long)Bn * LP * LP;  // 5,308,416
  float* xp    = ws;           // [B, LP, 256]  padded x_
  float* bufA  = xp + NLD;     // q_  -> att_v
  float* bufB  = bufA + NLL;   // scores/att -> y1 -> hv   (sized NLL >= NLD)
  float* bufC  = bufB + NLL;   // ln(att_v2)
  float* aent  = bufC + NLD;   // [B]
  float* aglob = aent + Bn;    // [B]
  float* msw2  = aglob + Bn;   // [B, 256]
  float* al    = msw2 + Bn * Dd;  // [B, LP]
  float* gc    = al + Bn * LP;    // [B, 256]
  float* lc    = gc + Bn * Dd;    // [B, 128]
  float* invn  = lc + Bn * IDIM;  // [99999]

  const dim3 blk(128);  // 4 waves per block for GEMMs

  // 1) x_ = concat(emb[x], pos_emb[pos]), zero-padded to LP rows
  k_embed<<<Bn * LP, 256, 0, stream>>>(x, pos, emb, pemb, xp);
  // 2) a_ent from x_[:, -1, :]
  k_alpha<<<Bn, 256, 0, stream>>>(xp, alpha_w, alpha_b, aent);
  // 3) q_ = relu(x_ @ mlp_w^T + mlp_b)     [18432 x 256]
  k_gemm32<1><<<dim3(1152, 1), blk, 0, stream>>>(
      xp, mlp_w, bufA, (Bn * LP) / 32, 8, Dd, Dd, Dd, Dd, 0, 0, 0, mlp_b, 1,
      nullptr, 1.0f);
  // 4) scores = q_ @ x_^T / 16   per batch [288 x 288]
  k_gemm32<1><<<dim3(21, Bn), blk, 0, stream>>>(
      bufA, xp, bufB, LP / 32, LP / 32, Dd, Dd, Dd, LP, (long long)LP * Dd,
      (long long)LP * Dd, (long long)LP * LP, nullptr, 0, nullptr, 0.0625f);
  // 5) att = entmax_bisect(scores, a_ent)  (in place, masked, padding zeroed)
  k_entmax_local<<<(Bn * LP) / 8, 256, 0, stream>>>(bufB, x, aent);
  // 6) att_v = att @ x_    per batch [288 x 256], K=288
  k_gemm32<0><<<dim3(18, Bn), blk, 0, stream>>>(
      bufB, xp, bufA, LP / 32, Dd / 32, LP, LP, Dd, Dd, (long long)LP * LP,
      (long long)LP * Dd, (long long)LP * Dd, nullptr, 0, nullptr, 1.0f);
  // 7) y1 = relu(att_v @ sa_w1^T + b1)
  k_gemm32<1><<<dim3(1152, 1), blk, 0, stream>>>(
      bufA, sa_w1_w, bufB, (Bn * LP) / 32, 8, Dd, Dd, Dd, Dd, 0, 0, 0, sa_w1_b,
      1, nullptr, 1.0f);
  // 8) t = y1 @ sa_w2^T + b2 + att_v ; then layernorm
  k_gemm32<1><<<dim3(1152, 1), blk, 0, stream>>>(
      bufB, sa_w2_w, bufC, (Bn * LP) / 32, 8, Dd, Dd, Dd, Dd, 0, 0, 0, sa_w2_b,
      0, bufA, 1.0f);
  k_ln<<<Bn * LP, 256, 0, stream>>>(bufC, ln_g, ln_b);
  // 9) a_glob from m_s = ln row 284
  k_alpha<<<Bn, 256, 0, stream>>>(bufC, alpha_w, alpha_b, aglob);
  // 10) hv = ln_out @ atten_w1 (no transpose)
  k_gemm32<0><<<dim3(1152, 1), blk, 0, stream>>>(
      bufC, atten_w1, bufB, (Bn * LP) / 32, 8, Dd, Dd, Dd, Dd, 0, 0, 0, nullptr,
      0, nullptr, 1.0f);
  // 11) msw2 = m_s @ atten_w2 ; al = relu(hv + msw2 + bias) . atten_w0
  k_msw2<<<Bn, 256, 0, stream>>>(bufC, atten_w2, msw2);
  k_al<<<(Bn * Lm1) / 8, 256, 0, stream>>>(bufB, msw2, atten_bias, atten_w0, al);
  // 12) aw = entmax_bisect(al, a_glob)  (in place)
  k_entmax_global<<<8, 256, 0, stream>>>(al, x, aglob);
  // 13) global_c = aw @ x_s
  k_globalc<<<Bn, 256, 0, stream>>>(al, xp, gc);
  // 14) l_c = normalize(selu(concat(gc, m_s) @ w_f^T + b_f))
  k_final<<<Bn, 128, 0, stream>>>(gc, bufC, w_f_w, w_f_b, lc);
  // 15) inv row norms of emb[1..99999]
  k_invnorm<<<12500, 256, 0, stream>>>(emb, invn);
  // 16) z = 20 * l_c @ l_emb^T   [64 x 99999]
  k_z<<<782, 256, 0, stream>>>(lc, emb, invn, (float*)d_out);
}